// MLP_GRAPH_GEN_55490977464357
// MI455X (gfx1250) — compile-verified
//
#include <hip/hip_runtime.h>
#include <math.h>

// ---------------------------------------------------------------------------
// Problem constants (from reference): N=12288 rows, F=768, H=512, O=256,
// keep top (KNN_K+1)=31 per row of the 12288x12288 cosine-sim matrix.
// ---------------------------------------------------------------------------
#define N_ROWS   12288
#define F_DIM    768
#define H_DIM    512
#define O_DIM    256
#define KNN_KEEP 31
#define EPSV     1e-12f

#define WAVES_PER_BLOCK 8   // 256 threads, wave32
#define TM 2                // 16-row tiles per wave (32 rows)
#define TN 4                // 16-col tiles per wave (64 cols)

typedef __attribute__((ext_vector_type(2))) float v2f;
typedef __attribute__((ext_vector_type(8))) float v8f;

// ---------------------------------------------------------------------------
// Generic Y = act(X @ W^T + b) with V_WMMA_F32_16X16X4_F32.
//   X: [M, K] row-major, W: [Nout, K] row-major, Y: [M, Nout].
// Each wave computes a 32x64 register tile (TM x TN 16x16 WMMA tiles).
// A (16x4 of X) and B (4x16 of W^T == rows of W) fragments are
// "2 consecutive floats per lane" per the CDNA5 f32 WMMA layouts:
//   lane 0..15  : row (base+lane),    K = k+0,k+1
//   lane 16..31 : row (base+lane-16), K = k+2,k+3
// C/D layout: VGPR i, lanes 0-15 -> (M=i, N=lane); lanes 16-31 -> (M=8+i).
// The K loop is software-pipelined: next-step fragments are loaded into
// registers before the current step's 8 independent WMMA chains issue, so
// global-load latency overlaps the matrix pipe instead of serializing it.
// ---------------------------------------------------------------------------
__global__ __launch_bounds__(256) void wmma_gemm_xwt(
    const float* __restrict__ X,
    const float* __restrict__ W,
    const float* __restrict__ bias,   // nullptr -> no bias
    float* __restrict__ Y,
    int M, int K, int Nout, int do_relu)
{
    const int wave = threadIdx.x >> 5;
    const int lane = threadIdx.x & 31;
    const int half = lane >> 4;       // 0: K=0..1, 1: K=2..3
    const int r    = lane & 15;       // row within 16-tile

    const int strips_n = Nout / (16 * TN);
    const long strip   = (long)blockIdx.x * WAVES_PER_BLOCK + wave;
    const int  mTile   = (int)(strip / strips_n);
    const int  nStrip  = (int)(strip % strips_n);
    if (mTile >= M / (16 * TM)) return;  // wave-uniform: EXEC stays all-1s

    const int mBase = mTile * (16 * TM);
    const int nBase = nStrip * (16 * TN);

    const float* arow[TM];
#pragma unroll
    for (int im = 0; im < TM; ++im)
        arow[im] = X + (size_t)(mBase + 16 * im + r) * K + half * 2;
    const float* brow[TN];
#pragma unroll
    for (int t = 0; t < TN; ++t)
        brow[t] = W + (size_t)(nBase + 16 * t + r) * K + half * 2;

    v8f acc[TM][TN];
#pragma unroll
    for (int im = 0; im < TM; ++im)
#pragma unroll
        for (int t = 0; t < TN; ++t)
            acc[im][t] = (v8f){0.f,0.f,0.f,0.f,0.f,0.f,0.f,0.f};

    // ---- software-pipelined K loop (step = 4) ----
    v2f a_cur[TM], b_cur[TN];
#pragma unroll
    for (int im = 0; im < TM; ++im) a_cur[im] = *(const v2f*)(arow[im]);
#pragma unroll
    for (int t = 0; t < TN; ++t)    b_cur[t]  = *(const v2f*)(brow[t]);

    for (int k = 4; k < K; k += 4) {
        v2f a_nxt[TM], b_nxt[TN];
#pragma unroll
        for (int im = 0; im < TM; ++im) a_nxt[im] = *(const v2f*)(arow[im] + k);
#pragma unroll
        for (int t = 0; t < TN; ++t)    b_nxt[t]  = *(const v2f*)(brow[t] + k);

#pragma unroll
        for (int im = 0; im < TM; ++im)
#pragma unroll
            for (int t = 0; t < TN; ++t)
                acc[im][t] = __builtin_amdgcn_wmma_f32_16x16x4_f32(
                    false, a_cur[im], false, b_cur[t], (short)0,
                    acc[im][t], false, false);

#pragma unroll
        for (int im = 0; im < TM; ++im) a_cur[im] = a_nxt[im];
#pragma unroll
        for (int t = 0; t < TN; ++t)    b_cur[t]  = b_nxt[t];
    }
    // tail step
#pragma unroll
    for (int im = 0; im < TM; ++im)
#pragma unroll
        for (int t = 0; t < TN; ++t)
            acc[im][t] = __builtin_amdgcn_wmma_f32_16x16x4_f32(
                false, a_cur[im], false, b_cur[t], (short)0,
                acc[im][t], false, false);

    // ---- epilogue: bias + optional relu, scatter per C/D layout ----
#pragma unroll
    for (int t = 0; t < TN; ++t) {
        const int col = nBase + 16 * t + r;
        const float bv = bias ? bias[col] : 0.0f;
#pragma unroll
        for (int im = 0; im < TM; ++im) {
#pragma unroll
            for (int i = 0; i < 8; ++i) {
                const int row = mBase + 16 * im + i + half * 8;
                float v = acc[im][t][i] + bv;
                if (do_relu) v = fmaxf(v, 0.0f);
                Y[(size_t)row * Nout + col] = v;
            }
        }
    }
}

// ---------------------------------------------------------------------------
// Row-wise L2 normalization: e = x / max(||x||_2, eps).  One block per row,
// blockDim.x == O_DIM == 256 (one element per thread).
// ---------------------------------------------------------------------------
__global__ __launch_bounds__(256) void row_l2_normalize(
    const float* __restrict__ Xin, float* __restrict__ E, int Kdim)
{
    __shared__ float red[256];
    const int row = blockIdx.x;
    const int tid = threadIdx.x;
    const float x = Xin[(size_t)row * Kdim + tid];
    red[tid] = x * x;
    __syncthreads();
    for (int s = 128; s > 0; s >>= 1) {
        if (tid < s) red[tid] += red[tid + s];
        __syncthreads();
    }
    const float nrm   = sqrtf(red[0]);
    const float scale = 1.0f / fmaxf(nrm, EPSV);
    E[(size_t)row * Kdim + tid] = x * scale;
}

// ---------------------------------------------------------------------------
// Per-row exact top-K threshold + relu mask, in place on the sim matrix.
// Map floats to order-preserving uint32, cache the row in LDS (48 KB),
// bit-build the largest threshold t with count(u >= t) >= KNN_KEEP (== the
// K-th largest value exactly), then write relu(x) where kept, 0 elsewhere.
// ---------------------------------------------------------------------------
__global__ __launch_bounds__(256) void topk_threshold_relu(
    float* __restrict__ S, int Ncols)
{
    __shared__ unsigned int srow[N_ROWS];   // 49152 B  (< 320 KB WGP LDS)
    __shared__ unsigned int wcnt[WAVES_PER_BLOCK];
    __shared__ unsigned int total;

    const int row = blockIdx.x;
    const int tid = threadIdx.x;
    float* rp = S + (size_t)row * Ncols;

    for (int j = tid; j < Ncols; j += 256) {
        unsigned int b = __float_as_uint(rp[j]);
        srow[j] = (b & 0x80000000u) ? ~b : (b | 0x80000000u);
    }
    __syncthreads();

    unsigned int thresh = 0u;
    for (int bit = 31; bit >= 0; --bit) {
        const unsigned int cand = thresh | (1u << bit);
        unsigned int c = 0;
        for (int j = tid; j < Ncols; j += 256)
            c += (srow[j] >= cand) ? 1u : 0u;
        for (int off = 16; off > 0; off >>= 1)
            c += (unsigned int)__shfl_xor((int)c, off, 32);
        if ((tid & 31) == 0) wcnt[tid >> 5] = c;
        __syncthreads();
        if (tid == 0) {
            unsigned int t2 = 0;
            for (int w = 0; w < WAVES_PER_BLOCK; ++w) t2 += wcnt[w];
            total = t2;
        }
        __syncthreads();
        if (total >= KNN_KEEP) thresh = cand;
        __syncthreads();   // protect wcnt/total before next iteration
    }

    for (int j = tid; j < Ncols; j += 256) {
        const unsigned int u = srow[j];
        const unsigned int b = (u & 0x80000000u) ? (u & 0x7FFFFFFFu) : ~u;
        const float f = __uint_as_float(b);
        rp[j] = (u >= thresh && f > 0.0f) ? f : 0.0f;
    }
}

// ---------------------------------------------------------------------------
// kernel_launch: fused pipeline, all on `stream`, graph-capture safe.
// d_ws layout (needs >= 48 MB):
//   x1 @ 0          (12288*512*4 = 25165824 B)
//   x2 @ 25165824   (25165824 B)
//   x3 @ 0          (12 MB, overwrites x1 after it is consumed)
//   e  @ 25165824   (12 MB, overwrites x2 after it is consumed)
// sim goes straight into d_out and is masked in place.
// ---------------------------------------------------------------------------
extern "C" void kernel_launch(void* const* d_in, const int* in_sizes, int n_in,
                              void* d_out, int out_size, void* d_ws, size_t ws_size,
                              hipStream_t stream)
{
    const float* feat = (const float*)d_in[0];
    const float* W1   = (const float*)d_in[1];
    const float* b1   = (const float*)d_in[2];
    const float* W2   = (const float*)d_in[3];
    const float* b2   = (const float*)d_in[4];
    const float* W3   = (const float*)d_in[5];
    const float* b3   = (const float*)d_in[6];
    float* out = (float*)d_out;

    char*  ws = (char*)d_ws;
    float* x1 = (float*)(ws + 0);
    float* x2 = (float*)(ws + 25165824);
    float* x3 = (float*)(ws + 0);
    float* e  = (float*)(ws + 25165824);

    const dim3 blk(256);
    auto gemm_blocks = [](int M, int Nout) {
        return (unsigned)(((size_t)(M / (16 * TM)) * (Nout / (16 * TN)))
                          / WAVES_PER_BLOCK);
    };

    // x1 = relu(feat @ W1^T + b1)
    wmma_gemm_xwt<<<gemm_blocks(N_ROWS, H_DIM), blk, 0, stream>>>(
        feat, W1, b1, x1, N_ROWS, F_DIM, H_DIM, 1);
    // x2 = relu(x1 @ W2^T + b2)
    wmma_gemm_xwt<<<gemm_blocks(N_ROWS, H_DIM), blk, 0, stream>>>(
        x1, W2, b2, x2, N_ROWS, H_DIM, H_DIM, 1);
    // x3 = x2 @ W3^T + b3
    wmma_gemm_xwt<<<gemm_blocks(N_ROWS, O_DIM), blk, 0, stream>>>(
        x2, W3, b3, x3, N_ROWS, H_DIM, O_DIM, 0);
    // e = l2normalize(x3)
    row_l2_normalize<<<N_ROWS, blk, 0, stream>>>(x3, e, O_DIM);
    // sim = e @ e^T  (E fits in L2: 12 MB << 192 MB)
    wmma_gemm_xwt<<<gemm_blocks(N_ROWS, N_ROWS), blk, 0, stream>>>(
        e, e, nullptr, out, N_ROWS, O_DIM, N_ROWS, 0);
    // per-row exact top-31 mask + relu, in place
    topk_threshold_relu<<<N_ROWS, blk, 0, stream>>>(out, N_ROWS);
}